// ModelMM_13357348290738
// MI455X (gfx1250) — compile-verified
//
#include <hip/hip_runtime.h>
#include <math.h>

// Problem constants (from reference)
constexpr int   kBS   = 64;
constexpr int   kL    = 100;
constexpr int   kD    = 64;
constexpr int   kL1   = 101;
constexpr int   kN    = 6464;   // BS*(L+1) columns
constexpr int   kRows = 6400;   // BS*L rows
constexpr float kNegInf = -10000.0f;

typedef __attribute__((ext_vector_type(2))) float v2f;
typedef __attribute__((ext_vector_type(8))) float v8f;

// ---------------------------------------------------------------------------
// Kernel 1: per-column debias/valid + per-(batch,column) duplicate mask
// grid = (ceil(kN/256), kBS), block = 256
// ---------------------------------------------------------------------------
__global__ void prep_kernel(const int* __restrict__ ids,
                            const float* __restrict__ pop,
                            const int* __restrict__ log_mask,
                            float* __restrict__ debias,
                            unsigned char* __restrict__ valid,
                            unsigned char* __restrict__ dup) {
    __shared__ int sids[kL1];
    const int b   = blockIdx.y;
    const int tid = threadIdx.x;
    if (tid < kL1) sids[tid] = ids[b * kL1 + tid];
    __syncthreads();

    const int k = blockIdx.x * blockDim.x + tid;
    if (k >= kN) return;

    const int myid = ids[k];
    bool d = false;
    #pragma unroll 4
    for (int i = 0; i < kL1; ++i) d = d || (sids[i] == myid);
    dup[b * kN + k] = d ? 1 : 0;

    if (b == 0) {
        debias[k] = logf(pop[myid]);
        const int kb = k / kL1, kj = k % kL1;
        valid[k] = (kj == kL) ? (unsigned char)1
                              : (unsigned char)(log_mask[kb * kL + kj] != 0);
    }
}

// ---------------------------------------------------------------------------
// Kernel 2: fused WMMA GEMM + masking + online logsumexp + target gather
// grid = kRows/16 = 400, block = 256 (8 wave32 waves)
// Column tiles are software-pipelined: next tile's B fragments + column
// metadata are loaded (one big clause, full MLP) before the current tile's
// 16-deep v_wmma_f32_16x16x4_f32 chain, which hides their latency.
// ---------------------------------------------------------------------------
__global__ __launch_bounds__(256) void fused_kernel(
    const float* __restrict__ prec_vec,
    const float* __restrict__ score_embs,
    const float* __restrict__ debias,
    const unsigned char* __restrict__ valid,
    const unsigned char* __restrict__ dup,
    const int* __restrict__ log_mask,
    float* __restrict__ row_nll,
    float* __restrict__ row_w) {

    const int tid  = threadIdx.x;
    const int wid  = tid >> 5;          // wave id 0..7
    const int lane = tid & 31;
    const int half = lane >> 4;         // 0: K%4 in {0,1}; 1: K%4 in {2,3}
    const int lq   = lane & 15;
    const int koff = 2 * half;

    const int row_base = blockIdx.x * 16;
    const int bA = row_base / kL;
    const int bB = (row_base + 15) / kL;

    // A fragments: lane's row = row_base + lq, K striped per 16x4 f32 layout
    const float* arow = prec_vec + (size_t)(row_base + lq) * kD;
    v2f afrag[16];
    #pragma unroll
    for (int s = 0; s < 16; ++s)
        afrag[s] = *(const v2f*)(arow + 4 * s + koff);

    // Per-row constants for the 8 rows this lane accumulates in C
    // C layout: VGPR r -> row (r + 8*half), column (col_base + lq)
    int  tg[8];
    bool selA[8];
    #pragma unroll
    for (int r = 0; r < 8; ++r) {
        const int g  = row_base + r + 8 * half;
        const int bb = g / kL;
        tg[r]   = g + bb + 1;           // = b*101 + j + 1
        selA[r] = (bb == bA);
    }

    float m[8], s[8], tval[8];
    #pragma unroll
    for (int r = 0; r < 8; ++r) { m[r] = -3.0e38f; s[r] = 0.0f; tval[r] = -3.0e38f; }

    const int NT = kN / 16;             // 404 column tiles; this wave: wid, wid+8, ...

    // ---- pipeline prologue: load first tile ----
    v2f   bf[16];
    float deb;
    int   colC;
    bool  cvb, dAv, dBv;
    {
        const int col = wid * 16 + lq;
        const float* erow = score_embs + (size_t)col * kD;
        #pragma unroll
        for (int s2 = 0; s2 < 16; ++s2)
            bf[s2] = *(const v2f*)(erow + 4 * s2 + koff);
        deb  = debias[col];
        cvb  = valid[col] != 0;
        dAv  = dup[bA * kN + col] != 0;
        dBv  = dup[bB * kN + col] != 0;
        colC = col;
    }

    for (int ct = wid; ct < NT; ct += 8) {
        const int ctn = ct + 8;

        // ---- issue next tile's loads before this tile's WMMA chain ----
        v2f   bn[16];
        float debN = 0.0f;
        int   colN = 0;
        bool  cvN = false, dAN = false, dBN = false;
        if (ctn < NT) {
            const int col = ctn * 16 + lq;
            const float* erow = score_embs + (size_t)col * kD;
            #pragma unroll
            for (int s2 = 0; s2 < 16; ++s2)
                bn[s2] = *(const v2f*)(erow + 4 * s2 + koff);
            debN = debias[col];
            cvN  = valid[col] != 0;
            dAN  = dup[bA * kN + col] != 0;
            dBN  = dup[bB * kN + col] != 0;
            colN = col;
        } else {
            #pragma unroll
            for (int s2 = 0; s2 < 16; ++s2) bn[s2] = bf[s2];  // keep defined
        }

        // ---- 16-deep f32 WMMA accumulation chain (K = 64) ----
        v8f c = {0.f, 0.f, 0.f, 0.f, 0.f, 0.f, 0.f, 0.f};
        #pragma unroll
        for (int ks = 0; ks < 16; ++ks) {
            c = __builtin_amdgcn_wmma_f32_16x16x4_f32(
                    /*neg_a=*/false, afrag[ks], /*neg_b=*/false, bf[ks],
                    /*c_mod=*/(short)0, c, /*reuse_a=*/false, /*reuse_b=*/false);
        }

        // ---- masking + branchless online logsumexp ----
        #pragma unroll
        for (int r = 0; r < 8; ++r) {
            float v = cvb ? (c[r] - deb) : kNegInf;
            const bool dd    = selA[r] ? dAv : dBv;
            const bool istgt = (colC == tg[r]);
            v = (dd && !istgt) ? kNegInf : v;
            tval[r] = istgt ? v : tval[r];
            const float mo = m[r];
            const float M  = fmaxf(mo, v);
            s[r] = s[r] * __expf(mo - M) + __expf(v - M);
            m[r] = M;
        }

        // ---- rotate pipeline buffers ----
        #pragma unroll
        for (int s2 = 0; s2 < 16; ++s2) bf[s2] = bn[s2];
        deb = debN; cvb = cvN; dAv = dAN; dBv = dBN; colC = colN;
    }

    // Butterfly reduce across the 16-lane group holding one row's columns
    #pragma unroll
    for (int off = 1; off < 16; off <<= 1) {
        #pragma unroll
        for (int r = 0; r < 8; ++r) {
            const float mo = __shfl_xor(m[r],    off, 32);
            const float so = __shfl_xor(s[r],    off, 32);
            const float to = __shfl_xor(tval[r], off, 32);
            const float M  = fmaxf(m[r], mo);
            s[r] = s[r] * __expf(m[r] - M) + so * __expf(mo - M);
            m[r] = M;
            tval[r] = fmaxf(tval[r], to);
        }
    }

    __shared__ float lm[8][16], ls[8][16], lt[8][16];
    if (lq == 0) {
        #pragma unroll
        for (int r = 0; r < 8; ++r) {
            const int rl = r + 8 * half;
            lm[wid][rl] = m[r];
            ls[wid][rl] = s[r];
            lt[wid][rl] = tval[r];
        }
    }
    __syncthreads();

    // Cross-wave merge (fixed order -> deterministic), final per-row NLL
    if (tid < 16) {
        float M = lm[0][tid], S = ls[0][tid], T = lt[0][tid];
        #pragma unroll
        for (int w = 1; w < 8; ++w) {
            const float mo = lm[w][tid], so = ls[w][tid];
            const float Mn = fmaxf(M, mo);
            S = S * __expf(M - Mn) + so * __expf(mo - Mn);
            M = Mn;
            T = fmaxf(T, lt[w][tid]);
        }
        const int g  = row_base + tid;
        const int bb = g / kL, jj = g - bb * kL;
        const float lse = M + logf(S);
        const float w   = (log_mask[bb * kL + jj] != 0) ? 1.0f : 0.0f;
        row_nll[g] = (lse - T) * w;
        row_w[g]   = w;
    }
}

// ---------------------------------------------------------------------------
// Kernel 3: deterministic fixed-order reduction to scalar loss
// ---------------------------------------------------------------------------
__global__ void reduce_kernel(const float* __restrict__ rn,
                              const float* __restrict__ rw,
                              float* __restrict__ out) {
    __shared__ float sn[256], sw[256];
    const int tid = threadIdx.x;
    float a = 0.0f, b = 0.0f;
    for (int g = tid; g < kRows; g += 256) { a += rn[g]; b += rw[g]; }
    sn[tid] = a; sw[tid] = b;
    __syncthreads();
    for (int st = 128; st > 0; st >>= 1) {
        if (tid < st) { sn[tid] += sn[tid + st]; sw[tid] += sw[tid + st]; }
        __syncthreads();
    }
    if (tid == 0) out[0] = sn[0] / sw[0];
}

// ---------------------------------------------------------------------------
extern "C" void kernel_launch(void* const* d_in, const int* in_sizes, int n_in,
                              void* d_out, int out_size, void* d_ws, size_t ws_size,
                              hipStream_t stream) {
    const float* prec  = (const float*)d_in[0];   // [6400,64]
    const float* embs  = (const float*)d_in[1];   // [6464,64]
    const float* pop   = (const float*)d_in[2];   // [100001]
    const int*   ids   = (const int*)d_in[3];     // [6464]
    const int*   lmask = (const int*)d_in[4];     // [64,100]
    float* out = (float*)d_out;

    char* ws = (char*)d_ws;
    float*         debias  = (float*)(ws + 0);               // 6464*4   = 25856
    unsigned char* valid   = (unsigned char*)(ws + 25856);   // 6464
    unsigned char* dup     = (unsigned char*)(ws + 32320);   // 64*6464 = 413696
    float*         row_nll = (float*)(ws + 446016);          // 6400*4
    float*         row_w   = (float*)(ws + 471616);          // 6400*4 (ends 497216)

    dim3 g1((kN + 255) / 256, kBS);
    prep_kernel<<<g1, 256, 0, stream>>>(ids, pop, lmask, debias, valid, dup);
    fused_kernel<<<kRows / 16, 256, 0, stream>>>(prec, embs, debias, valid, dup,
                                                 lmask, row_nll, row_w);
    reduce_kernel<<<1, 256, 0, stream>>>(row_nll, row_w, out);
}